// Net_82901458747985
// MI455X (gfx1250) — compile-verified
//
#include <hip/hip_runtime.h>
#include <math.h>

#define IN_CH 14
#define HEADS 8
#define HID   128
#define OUTC  4
#define FDIM  (HEADS*HID)   // 1024
#define NEG_SLOPE 0.2f

typedef __attribute__((ext_vector_type(2))) float v2f;
typedef __attribute__((ext_vector_type(8))) float v8f;

// ---------------- helpers ----------------

__device__ __forceinline__ void edge_sd(const long long* __restrict__ ei, int e, int E,
                                        int& s, int& d) {
  if (e < E) { s = (int)ei[e]; d = (int)ei[E + e]; }
  else       { s = d = e - E; }              // self-loops appended
}

// float atomic max via sign-split int trick (valid for non-NaN IEEE)
__device__ __forceinline__ void atomicMaxF(float* addr, float val) {
  if (val >= 0.f) atomicMax((int*)addr, __float_as_int(val));
  else            atomicMin((unsigned int*)addr, (unsigned int)__float_as_int(val));
}

__global__ void fill_kernel(float* p, float v, int n) {
  int i = blockIdx.x * blockDim.x + threadIdx.x;
  if (i < n) p[i] = v;
}

// ---------------- layer matmuls ----------------

// H = x @ W1   (N x 14) @ (14 x 1024), K tiny -> plain VALU
__global__ void gemm1_kernel(const float* __restrict__ x, const float* __restrict__ W,
                             float* __restrict__ H, int n) {
  int i = blockIdx.x * blockDim.x + threadIdx.x;
  if (i >= n * FDIM) return;
  int node = i >> 10, col = i & (FDIM - 1);
  const float* xr = x + node * IN_CH;
  float s = 0.f;
#pragma unroll
  for (int k = 0; k < IN_CH; ++k) s = fmaf(xr[k], W[k * FDIM + col], s);
  H[i] = s;
}

#define ASTRIDE (32 * 17)    // padded A tile (32 rows x 16 K, stride 17)
#define BSTRIDE (16 * 64)    // B tile (16 K x 64 cols)

// C = A @ B with V_WMMA_F32_16X16X4_F32.
// Double-buffered LDS staging on ASYNCcnt: while chunk i is consumed by WMMA,
// chunk i+1 streams in via global_load_async_to_lds; s_wait_asynccnt 0x2
// releases chunk i (async loads complete in order, 2 issued per wave per chunk).
// Block: 256 threads = 8 waves arranged 2(M) x 4(N) -> 32 x 64 C region.
__global__ __launch_bounds__(256) void gemm2_wmma_kernel(const float* __restrict__ A,
                                                         const float* __restrict__ B,
                                                         float* __restrict__ C,
                                                         int M, int K, int Nc, int mblocks) {
  __shared__ float sA[2 * ASTRIDE];
  __shared__ float sB[2 * BSTRIDE];

  const int tid  = threadIdx.x;
  const int wid  = tid >> 5;
  const int lane = tid & 31;
  const int half = lane >> 4;        // K-pair / M-half selector
  const int l16  = lane & 15;
  const int bm = blockIdx.x % mblocks;
  const int bn = blockIdx.x / mblocks;
  const int rowBase = bm * 32;
  const int colBase = bn * 64;
  const int mi = wid >> 2, ni = wid & 3;
  const int row0 = rowBase + mi * 16;
  const int col0 = colBase + ni * 16;

  // per-thread async copy assignment
  const int ar = tid >> 3;            // A tile row 0..31
  const int ak = (tid & 7) * 2;       // A K offset within chunk (b64 = 2 floats)
  const int br = tid >> 4;            // B tile K row 0..15
  const int bc = (tid & 15) * 4;      // B col group (b128 = 4 floats)

  int arow = rowBase + ar;
  if (arow >= M) arow = M - 1;        // clamp: duplicate row, never stored
  const float* gA = A + (size_t)arow * K + ak;          // chunk-0 addresses
  const float* gB = B + (size_t)br * Nc + colBase + bc;

  const unsigned ldsA0 = (unsigned)(uintptr_t)(&sA[ar * 17 + ak]);
  const unsigned ldsB0 = (unsigned)(uintptr_t)(&sB[br * 64 + bc]);

  // prologue: chunk 0 -> buffer 0
  asm volatile("global_load_async_to_lds_b64 %0, %1, off"
               :: "v"(ldsA0), "v"(gA) : "memory");
  asm volatile("global_load_async_to_lds_b128 %0, %1, off"
               :: "v"(ldsB0), "v"(gB) : "memory");

  v8f acc = {0.f, 0.f, 0.f, 0.f, 0.f, 0.f, 0.f, 0.f};
  int buf = 0;

  for (int k0 = 0; k0 < K; k0 += 16) {
    if (k0 + 16 < K) {
      // prefetch next chunk into the other buffer
      const unsigned ldsAn = ldsA0 + (unsigned)((buf ^ 1) * ASTRIDE * 4);
      const unsigned ldsBn = ldsB0 + (unsigned)((buf ^ 1) * BSTRIDE * 4);
      const float* gAn = gA + (k0 + 16);
      const float* gBn = gB + (size_t)(k0 + 16) * Nc;
      asm volatile("global_load_async_to_lds_b64 %0, %1, off"
                   :: "v"(ldsAn), "v"(gAn) : "memory");
      asm volatile("global_load_async_to_lds_b128 %0, %1, off"
                   :: "v"(ldsBn), "v"(gBn) : "memory");
      asm volatile("s_wait_asynccnt 0x2" ::: "memory");  // current chunk resident
    } else {
      asm volatile("s_wait_asynccnt 0x0" ::: "memory");
    }
    __syncthreads();

    const int abase = buf * ASTRIDE;
    const int bbase = buf * BSTRIDE;
#pragma unroll
    for (int kk = 0; kk < 4; ++kk) {
      const int kb = kk * 4 + 2 * half;
      v2f a, b;
      a.x = sA[abase + (mi * 16 + l16) * 17 + kb];
      a.y = sA[abase + (mi * 16 + l16) * 17 + kb + 1];
      b.x = sB[bbase + kb * 64 + ni * 16 + l16];
      b.y = sB[bbase + (kb + 1) * 64 + ni * 16 + l16];
      acc = __builtin_amdgcn_wmma_f32_16x16x4_f32(false, a, false, b, (short)0, acc,
                                                  false, false);
    }
    __syncthreads();   // protect buffer before it is overwritten next iteration
    buf ^= 1;
  }

  if (row0 < M) {   // C/D layout: VGPR r holds M=r (lanes 0-15) and M=8+r (lanes 16-31)
    float* Cp = C + (size_t)(row0 + half * 8) * Nc + col0 + l16;
#pragma unroll
    for (int r = 0; r < 8; ++r) Cp[(size_t)r * Nc] = acc[r];
  }
}

// O = H @ W3  (N x 1024) @ (1024 x 4): one wave per node, lane-strided K reduce
__global__ void gemm3_kernel(const float* __restrict__ H, const float* __restrict__ W,
                             float* __restrict__ O, int n) {
  int w = (blockIdx.x * blockDim.x + threadIdx.x) >> 5;
  int lane = threadIdx.x & 31;
  if (w >= n) return;
  const float* hp = H + (size_t)w * FDIM;
  float a0 = 0.f, a1 = 0.f, a2 = 0.f, a3 = 0.f;
  for (int k = lane; k < FDIM; k += 32) {
    float v = hp[k];
    const float* wr = W + k * OUTC;
    a0 = fmaf(v, wr[0], a0);
    a1 = fmaf(v, wr[1], a1);
    a2 = fmaf(v, wr[2], a2);
    a3 = fmaf(v, wr[3], a3);
  }
#pragma unroll
  for (int off = 16; off > 0; off >>= 1) {
    a0 += __shfl_down(a0, off);
    a1 += __shfl_down(a1, off);
    a2 += __shfl_down(a2, off);
    a3 += __shfl_down(a3, off);
  }
  if (lane == 0) {
    float* o = O + (size_t)w * OUTC;
    o[0] = a0; o[1] = a1; o[2] = a2; o[3] = a3;
  }
}

// ---------------- attention ----------------

// one wave per (node, head): al_s = <h, a_src>, al_d = <h, a_dst>
__global__ void attn_logits_kernel(const float* __restrict__ H,
                                   const float* __restrict__ asrc,
                                   const float* __restrict__ adst,
                                   float* __restrict__ als, float* __restrict__ ald,
                                   int n, int heads, int ch) {
  int w = (blockIdx.x * blockDim.x + threadIdx.x) >> 5;
  int lane = threadIdx.x & 31;
  if (w >= n * heads) return;
  int hd = w % heads;
  const float* hp = H + (size_t)w * ch;   // [node][head][ch] contiguous
  float s = 0.f, d = 0.f;
  for (int c = lane; c < ch; c += 32) {
    float v = hp[c];
    s = fmaf(v, asrc[hd * ch + c], s);
    d = fmaf(v, adst[hd * ch + c], d);
  }
#pragma unroll
  for (int off = 16; off > 0; off >>= 1) {
    s += __shfl_down(s, off);
    d += __shfl_down(d, off);
  }
  if (lane == 0) { als[w] = s; ald[w] = d; }
}

// heads==1 / ch==4 variant: one thread per node
__global__ void attn3_kernel(const float* __restrict__ H3, const float* __restrict__ as3,
                             const float* __restrict__ ad3, float* __restrict__ als,
                             float* __restrict__ ald, int n) {
  int i = blockIdx.x * blockDim.x + threadIdx.x;
  if (i >= n) return;
  const float* h = H3 + (size_t)i * OUTC;
  float s = 0.f, d = 0.f;
#pragma unroll
  for (int c = 0; c < OUTC; ++c) {
    s = fmaf(h[c], as3[c], s);
    d = fmaf(h[c], ad3[c], d);
  }
  als[i] = s; ald[i] = d;
}

// pass 1: leaky-relu logit per (edge,head), store, segment-max into m[dst]
__global__ void edge_max_kernel(const long long* __restrict__ ei,
                                const float* __restrict__ als, const float* __restrict__ ald,
                                float* __restrict__ m, float* __restrict__ el,
                                int E, int n, int heads) {
  int t = blockIdx.x * blockDim.x + threadIdx.x;
  if (t >= (E + n) * heads) return;
  int e = t / heads, hd = t - e * heads;
  int s, d; edge_sd(ei, e, E, s, d);
  float v = als[s * heads + hd] + ald[d * heads + hd];
  v = v > 0.f ? v : NEG_SLOPE * v;
  el[t] = v;
  atomicMaxF(&m[d * heads + hd], v);
}

// pass 2: ex = exp(v - m[dst]); store; segment-sum into z[dst]
__global__ void edge_expsum_kernel(const long long* __restrict__ ei,
                                   float* __restrict__ el, const float* __restrict__ m,
                                   float* __restrict__ z, int E, int n, int heads) {
  int t = blockIdx.x * blockDim.x + threadIdx.x;
  if (t >= (E + n) * heads) return;
  int e = t / heads, hd = t - e * heads;
  int s, d; edge_sd(ei, e, E, s, d);
  float ex = __expf(el[t] - m[d * heads + hd]);
  el[t] = ex;
  atomicAdd(&z[d * heads + hd], ex);
}

// pass 3: one wave per edge; agg[dst] += alpha * h[src], float4 gathers
__global__ void aggregate_kernel(const long long* __restrict__ ei,
                                 const float* __restrict__ H, const float* __restrict__ el,
                                 const float* __restrict__ z, float* __restrict__ agg,
                                 int E, int n, int heads, int ch) {
  int w = (blockIdx.x * blockDim.x + threadIdx.x) >> 5;
  int lane = threadIdx.x & 31;
  if (w >= E + n) return;
  int s, d; edge_sd(ei, w, E, s, d);
  int F = heads * ch;
  const float4* hs = (const float4*)(H + (size_t)s * F);
  float* ad = agg + (size_t)d * F;
  for (int i4 = lane; i4 * 4 < F; i4 += 32) {
    int i = i4 * 4;                      // 4 consecutive channels, same head (4 | ch)
    int hd = i / ch;
    float a = el[(size_t)w * heads + hd] / (z[d * heads + hd] + 1e-16f);
    float4 v = hs[i4];
    atomicAdd(&ad[i + 0], a * v.x);
    atomicAdd(&ad[i + 1], a * v.y);
    atomicAdd(&ad[i + 2], a * v.z);
    atomicAdd(&ad[i + 3], a * v.w);
  }
}

// heads==1 / ch==4 variant: one thread per edge
__global__ void aggregate3_kernel(const long long* __restrict__ ei,
                                  const float* __restrict__ H3, const float* __restrict__ el,
                                  const float* __restrict__ z, float* __restrict__ agg,
                                  int E, int n) {
  int e = blockIdx.x * blockDim.x + threadIdx.x;
  if (e >= E + n) return;
  int s, d; edge_sd(ei, e, E, s, d);
  float a = el[e] / (z[d] + 1e-16f);
#pragma unroll
  for (int c = 0; c < OUTC; ++c)
    atomicAdd(&agg[(size_t)d * OUTC + c], a * H3[(size_t)s * OUTC + c]);
}

// agg = relu(agg + bias)   (F must be a power of two)
__global__ void bias_relu_kernel(float* __restrict__ p, const float* __restrict__ b,
                                 int total, int Fmask) {
  int i = blockIdx.x * blockDim.x + threadIdx.x;
  if (i >= total) return;
  float v = p[i] + b[i & Fmask];
  p[i] = v > 0.f ? v : 0.f;
}

__global__ void final_kernel(const float* __restrict__ agg, const float* __restrict__ b,
                             float* __restrict__ out, int n) {
  int i = blockIdx.x * blockDim.x + threadIdx.x;
  if (i >= n * OUTC) return;
  float v = agg[i] + b[i & (OUTC - 1)];
  out[i] = v > 0.f ? v : 0.f;
}

// ---------------- driver ----------------

extern "C" void kernel_launch(void* const* d_in, const int* in_sizes, int n_in,
                              void* d_out, int out_size, void* d_ws, size_t ws_size,
                              hipStream_t stream) {
  const float*     x      = (const float*)d_in[0];
  const long long* ei     = (const long long*)d_in[1];
  const float*     W1     = (const float*)d_in[2];
  const float*     asrc1  = (const float*)d_in[3];
  const float*     adst1  = (const float*)d_in[4];
  const float*     b1     = (const float*)d_in[5];
  const float*     W2     = (const float*)d_in[6];
  const float*     asrc2  = (const float*)d_in[7];
  const float*     adst2  = (const float*)d_in[8];
  const float*     b2     = (const float*)d_in[9];
  const float*     W3     = (const float*)d_in[10];
  const float*     asrc3  = (const float*)d_in[11];
  const float*     adst3  = (const float*)d_in[12];
  const float*     b3     = (const float*)d_in[13];
  float* out = (float*)d_out;

  const int N  = in_sizes[0] / IN_CH;   // 10000
  const int E  = in_sizes[1] / 2;       // 160000
  const int E2 = E + N;                 // with self-loops

  // workspace layout (floats)
  float* ws = (float*)d_ws;
  size_t o = 0;
  float* bufA   = ws + o; o += (size_t)N * FDIM;   // h (layer transform)
  float* bufB   = ws + o; o += (size_t)N * FDIM;   // aggregated output / next input
  float* als    = ws + o; o += (size_t)N * HEADS;
  float* ald    = ws + o; o += (size_t)N * HEADS;
  float* mbuf   = ws + o; o += (size_t)N * HEADS;
  float* zbuf   = ws + o; o += (size_t)N * HEADS;
  float* alpha  = ws + o; o += (size_t)E2 * HEADS;
  float* h3     = ws + o; o += (size_t)N * OUTC;
  float* agg3   = ws + o; o += (size_t)N * OUTC;
  float* als3   = ws + o; o += (size_t)N;
  float* ald3   = ws + o; o += (size_t)N;
  float* m3     = ws + o; o += (size_t)N;
  float* z3     = ws + o; o += (size_t)N;

  const int BT = 256;
  auto blocks = [](long long t, int b) { return (int)((t + b - 1) / b); };

  const int mblocks = (N + 31) / 32;            // 313
  const int nblocks = FDIM / 64;                // 16
  const int gemm_grid = mblocks * nblocks;

  // ---------------- layer 1 ----------------
  gemm1_kernel<<<blocks((long long)N * FDIM, BT), BT, 0, stream>>>(x, W1, bufA, N);
  attn_logits_kernel<<<blocks((long long)N * HEADS * 32, BT), BT, 0, stream>>>(
      bufA, asrc1, adst1, als, ald, N, HEADS, HID);
  fill_kernel<<<blocks(N * HEADS, BT), BT, 0, stream>>>(mbuf, -1e30f, N * HEADS);
  fill_kernel<<<blocks(N * HEADS, BT), BT, 0, stream>>>(zbuf, 0.f, N * HEADS);
  fill_kernel<<<blocks((long long)N * FDIM, BT), BT, 0, stream>>>(bufB, 0.f, N * FDIM);
  edge_max_kernel<<<blocks((long long)E2 * HEADS, BT), BT, 0, stream>>>(
      ei, als, ald, mbuf, alpha, E, N, HEADS);
  edge_expsum_kernel<<<blocks((long long)E2 * HEADS, BT), BT, 0, stream>>>(
      ei, alpha, mbuf, zbuf, E, N, HEADS);
  aggregate_kernel<<<blocks((long long)E2 * 32, BT), BT, 0, stream>>>(
      ei, bufA, alpha, zbuf, bufB, E, N, HEADS, HID);
  bias_relu_kernel<<<blocks((long long)N * FDIM, BT), BT, 0, stream>>>(
      bufB, b1, N * FDIM, FDIM - 1);

  // ---------------- layer 2 ----------------
  gemm2_wmma_kernel<<<gemm_grid, BT, 0, stream>>>(bufB, W2, bufA, N, FDIM, FDIM, mblocks);
  attn_logits_kernel<<<blocks((long long)N * HEADS * 32, BT), BT, 0, stream>>>(
      bufA, asrc2, adst2, als, ald, N, HEADS, HID);
  fill_kernel<<<blocks(N * HEADS, BT), BT, 0, stream>>>(mbuf, -1e30f, N * HEADS);
  fill_kernel<<<blocks(N * HEADS, BT), BT, 0, stream>>>(zbuf, 0.f, N * HEADS);
  fill_kernel<<<blocks((long long)N * FDIM, BT), BT, 0, stream>>>(bufB, 0.f, N * FDIM);
  edge_max_kernel<<<blocks((long long)E2 * HEADS, BT), BT, 0, stream>>>(
      ei, als, ald, mbuf, alpha, E, N, HEADS);
  edge_expsum_kernel<<<blocks((long long)E2 * HEADS, BT), BT, 0, stream>>>(
      ei, alpha, mbuf, zbuf, E, N, HEADS);
  aggregate_kernel<<<blocks((long long)E2 * 32, BT), BT, 0, stream>>>(
      ei, bufA, alpha, zbuf, bufB, E, N, HEADS, HID);
  bias_relu_kernel<<<blocks((long long)N * FDIM, BT), BT, 0, stream>>>(
      bufB, b2, N * FDIM, FDIM - 1);

  // ---------------- layer 3 (heads=1, out=4, concat=False) ----------------
  gemm3_kernel<<<blocks((long long)N * 32, BT), BT, 0, stream>>>(bufB, W3, h3, N);
  attn3_kernel<<<blocks(N, BT), BT, 0, stream>>>(h3, asrc3, adst3, als3, ald3, N);
  fill_kernel<<<blocks(N, BT), BT, 0, stream>>>(m3, -1e30f, N);
  fill_kernel<<<blocks(N, BT), BT, 0, stream>>>(z3, 0.f, N);
  fill_kernel<<<blocks(N * OUTC, BT), BT, 0, stream>>>(agg3, 0.f, N * OUTC);
  edge_max_kernel<<<blocks(E2, BT), BT, 0, stream>>>(ei, als3, ald3, m3, alpha, E, N, 1);
  edge_expsum_kernel<<<blocks(E2, BT), BT, 0, stream>>>(ei, alpha, m3, z3, E, N, 1);
  aggregate3_kernel<<<blocks(E2, BT), BT, 0, stream>>>(ei, h3, alpha, z3, agg3, E, N);
  final_kernel<<<blocks(N * OUTC, BT), BT, 0, stream>>>(agg3, b3, out, N);
}